// GD_257698038299
// MI455X (gfx1250) — compile-verified
//
#include <hip/hip_runtime.h>
#include <stdint.h>
#include <math.h>

#ifndef __has_builtin
#define __has_builtin(x) 0
#endif

// Problem constants (from reference)
#define R_    150
#define A_    6
#define S_    3
#define P_    32768
#define B_    8
#define K_RAS (R_ * A_ * S_)   // 2700
#define K_TOT (K_RAS + B_)     // 2708
#define TPB   256
#define PTS   2                 // points per thread in main kernel
#define MAX_CHUNK 2720          // >= K_TOT, float2 => 21,760 B LDS (of 320 KB/WGP)

typedef unsigned int v4u_t __attribute__((ext_vector_type(4)));
typedef int          v8i_t __attribute__((ext_vector_type(8)));
typedef int          v4i_t __attribute__((ext_vector_type(4)));

// ---------------------------------------------------------------------------
// Kernel 0: build packed (mu, w) peak table.
//   k < 2700 : mu = (isnan(shift)?0:shift) + offset - re_ref ; w = mask*softmax
//   k >= 2700: background peak, mu = bg + side_off - re_ref ; w = 1
// The common factor half/pi cancels under final normalization and is dropped.
// ---------------------------------------------------------------------------
__global__ void GD_prep_peaks(const float* __restrict__ shift,
                              const float* __restrict__ a,
                              const float* __restrict__ off,
                              const float* __restrict__ side_off,
                              const float* __restrict__ re_ref,
                              const float* __restrict__ bg,
                              float2* __restrict__ peaks) {
    int k = blockIdx.x * blockDim.x + threadIdx.x;
    if (k >= K_TOT) return;
    float re = re_ref[0];
    if (k < K_RAS) {
        int r = k / (A_ * S_);
        int s = k % S_;
        float a0 = a[r * S_ + 0], a1 = a[r * S_ + 1], a2 = a[r * S_ + 2];
        float m  = fmaxf(a0, fmaxf(a1, a2));
        float e0 = expf(a0 - m), e1 = expf(a1 - m), e2 = expf(a2 - m);
        float sel  = (s == 0) ? e0 : ((s == 1) ? e1 : e2);
        float soft = sel / (e0 + e1 + e2);
        float sh   = shift[k];
        bool  valid = !(sh != sh);   // !isnan
        float mu = (valid ? sh : 0.0f) + off[k] - re;
        peaks[k] = make_float2(mu, valid ? soft : 0.0f);
    } else {
        int i = k - K_RAS;
        peaks[k] = make_float2(bg[i] + side_off[i] - re, 1.0f);
    }
}

// ---------------------------------------------------------------------------
// Kernel 1: main accumulation. grid = (P/(TPB*PTS), KSPLIT).
// Stages this block's peak chunk into LDS with the CDNA5 Tensor Data Mover
// (TENSOR_LOAD_TO_LDS, TENSORcnt) -- one DMA per block, issued by wave 0 --
// then evaluates  acc[p] += w_k / ((x_p - mu_k)^2 + h^2)  with v_rcp_f32.
// TRANS-pipe bound: 3 VALU + 1 rcp per term.
// ---------------------------------------------------------------------------
__global__ void GD_lorentz_accum(const float* __restrict__ ppm,
                                 const float2* __restrict__ peaks,
                                 const float* __restrict__ bptr,
                                 float* __restrict__ partial,
                                 int chunk_len) {
    __shared__ float2 s_peaks[MAX_CHUNK];
    const int tid = threadIdx.x;
    const int ks  = blockIdx.y;
    const int chunk_start = ks * chunk_len;
    int chunkN = K_TOT - chunk_start;
    if (chunkN > chunk_len) chunkN = chunk_len;

#if __has_builtin(__builtin_amdgcn_tensor_load_to_lds)
    // --- TDM path: one tensor DMA moves the whole (mu,w) chunk to LDS. ---
    if (tid < 32) {   // single wave issues; TDM ignores EXEC, branch is scalar
        uint32_t lds_base = (uint32_t)(uintptr_t)(&s_peaks[0]); // flat lo32 == LDS addr
        uint64_t gaddr    = (uint64_t)(uintptr_t)(peaks + chunk_start);
        uint32_t nElem    = (uint32_t)chunkN * 2u;              // f32 elements (<= 5416)

        // D# group0: [1:0]=count=1 | lds_addr | global_addr[56:0] | type=2
        v4u_t g0;
        g0.x = 1u;
        g0.y = lds_base;
        g0.z = (uint32_t)gaddr;
        g0.w = ((uint32_t)(gaddr >> 32) & 0x01FFFFFFu) | (2u << 30);

        // D# group1: data_size=4B; tensor_dim0=nElem; tensor_dim1=1;
        //            tile_dim0=nElem; tile_dim1=1; stride0=nElem; mask=0
        v8i_t g1;
        g1[0] = (int)(2u << 16);                    // [17:16] data_size=2 (4B)
        g1[1] = (int)((nElem & 0xFFFFu) << 16);     // tensor_dim0[15:0] @ bits 63:48
        g1[2] = (int)((nElem >> 16) | (1u << 16));  // tensor_dim0[31:16] | tensor_dim1[15:0]=1
        g1[3] = (int)(nElem << 16);                 // tensor_dim1[31:16]=0 | tile_dim0=nElem
        g1[4] = 1;                                  // tile_dim1=1 ; tile_dim2=0
        g1[5] = (int)nElem;                         // tensor_dim0_stride[31:0]
        g1[6] = 0;                                  // stride0 hi | stride1 lo
        g1[7] = 0;                                  // stride1 hi

        v4i_t z4 = {0, 0, 0, 0};
#if __clang_major__ >= 23
        v8i_t z8 = {0, 0, 0, 0, 0, 0, 0, 0};
        __builtin_amdgcn_tensor_load_to_lds(g0, g1, z4, z4, z8, 0);
#else
        __builtin_amdgcn_tensor_load_to_lds(g0, g1, z4, z4, 0);
#endif
        __builtin_amdgcn_s_wait_tensorcnt(0);
    }
#else
    // --- Fallback: per-lane CDNA5 async global->LDS copies (ASYNCcnt). ---
    for (int i = tid; i < chunkN; i += TPB) {
        uint32_t lds = (uint32_t)(uintptr_t)(&s_peaks[i]);
        uint64_t ga  = (uint64_t)(uintptr_t)(peaks + chunk_start + i);
        asm volatile("global_load_async_to_lds_b64 %0, %1, off"
                     :: "v"(lds), "v"(ga) : "memory");
    }
    asm volatile("s_wait_asynccnt 0x0" ::: "memory");
#endif
    __syncthreads();

    float b0   = bptr[0];
    float b2   = fmaxf(b0 * b0, 0.001f);
    float half = 0.5f * b2;
    float h2   = half * half;

    const int p0 = blockIdx.x * (TPB * PTS) + tid;
    const int p1 = p0 + TPB;
    const float x0 = ppm[p0];
    const float x1 = ppm[p1];

    float acc0 = 0.0f, acc1 = 0.0f;
#pragma unroll 4
    for (int k = 0; k < chunkN; ++k) {
        float2 mw = s_peaks[k];            // ds_load_b64, broadcast (uniform addr)
        float d0 = x0 - mw.x;
        float d1 = x1 - mw.x;
        acc0 = fmaf(mw.y, __builtin_amdgcn_rcpf(fmaf(d0, d0, h2)), acc0);
        acc1 = fmaf(mw.y, __builtin_amdgcn_rcpf(fmaf(d1, d1, h2)), acc1);
    }
    float* out = partial + (size_t)ks * P_;
    out[p0] = acc0;
    out[p1] = acc1;
}

// ---------------------------------------------------------------------------
// Kernel 2: combine KSPLIT partials (fixed order -> deterministic), write
// unnormalized spec to d_out, fixed-tree block reduction -> blocksums.
// ---------------------------------------------------------------------------
__global__ void GD_combine(const float* __restrict__ partial,
                           float* __restrict__ spec,
                           float* __restrict__ blocksums,
                           int ksplit) {
    __shared__ float red[TPB];
    const int tid = threadIdx.x;
    const int p   = blockIdx.x * TPB + tid;
    float s = 0.0f;
    for (int i = 0; i < ksplit; ++i)
        s += partial[(size_t)i * P_ + p];
    spec[p]  = s;
    red[tid] = s;
    __syncthreads();
    for (int st = TPB / 2; st > 0; st >>= 1) {
        if (tid < st) red[tid] += red[tid + st];
        __syncthreads();
    }
    if (tid == 0) blocksums[blockIdx.x] = red[0];
}

// ---------------------------------------------------------------------------
// Kernel 3: every block sums the 128 block sums in fixed order (uniform ->
// scalar loads) and normalizes its slice of d_out.
// ---------------------------------------------------------------------------
__global__ void GD_normalize(float* __restrict__ spec,
                             const float* __restrict__ blocksums,
                             int nblocks) {
    float total = 0.0f;
    for (int i = 0; i < nblocks; ++i) total += blocksums[i];
    const int p = blockIdx.x * TPB + threadIdx.x;
    spec[p] = spec[p] / total;
}

// ---------------------------------------------------------------------------
extern "C" void kernel_launch(void* const* d_in, const int* in_sizes, int n_in,
                              void* d_out, int out_size, void* d_ws, size_t ws_size,
                              hipStream_t stream) {
    (void)in_sizes; (void)n_in; (void)out_size;

    const float* ppm   = (const float*)d_in[0];
    const float* shift = (const float*)d_in[1];
    const float* a     = (const float*)d_in[2];
    const float* off   = (const float*)d_in[3];
    const float* side  = (const float*)d_in[4];
    const float* reref = (const float*)d_in[5];
    const float* b     = (const float*)d_in[6];
    const float* bg    = (const float*)d_in[7];
    float* out = (float*)d_out;

    // Workspace layout: [peaks float2[K_TOT] | pad to 32KB] [partials] [blocksums]
    char*   ws       = (char*)d_ws;
    float2* peaks    = (float2*)ws;
    const size_t part_off = 32768;

    int ksplit = 16;  // depends only on constant ws_size -> deterministic
    while (ksplit > 1 &&
           part_off + (size_t)ksplit * P_ * sizeof(float) + 1024 > ws_size)
        ksplit >>= 1;

    float* partial   = (float*)(ws + part_off);
    float* blocksums = partial + (size_t)ksplit * P_;
    const int chunk_len = (K_TOT + ksplit - 1) / ksplit;
    const int nb2 = P_ / TPB;  // 128

    GD_prep_peaks<<<(K_TOT + TPB - 1) / TPB, TPB, 0, stream>>>(
        shift, a, off, side, reref, bg, peaks);

    dim3 grid1(P_ / (TPB * PTS), ksplit);
    GD_lorentz_accum<<<grid1, TPB, 0, stream>>>(ppm, peaks, b, partial, chunk_len);

    GD_combine<<<nb2, TPB, 0, stream>>>(partial, out, blocksums, ksplit);
    GD_normalize<<<nb2, TPB, 0, stream>>>(out, blocksums, nb2);
}